// MVGRLEncoder_13589276525118
// MI455X (gfx1250) — compile-verified
//
#include <hip/hip_runtime.h>
#include <hip/hip_bf16.h>
#include <stdint.h>

#define N_NODES  4096
#define E_EDGES  65536
#define BAG      16
#define BF_DIM   8192
#define CIN      128
#define HID      512
#define G_GRAPHS 64
#define ALPHA_F  0.2f
#define BETA_F   0.8f
#define N_ITERS  64

typedef float v2f __attribute__((ext_vector_type(2)));
typedef float v8f __attribute__((ext_vector_type(8)));

// ---------------------------------------------------------------- utilities
__global__ void k_fill_f(float* p, long n, float v) {
  long t = (long)blockIdx.x * blockDim.x + threadIdx.x;
  if (t < n) p[t] = v;
}
__global__ void k_fill_i(int* p, long n, int v) {
  long t = (long)blockIdx.x * blockDim.x + threadIdx.x;
  if (t < n) p[t] = v;
}

// ------------------------------------------------------------- embedding bag
__global__ void k_embed(const float* __restrict__ cb, const int* __restrict__ idx,
                        float* __restrict__ x) {
  int i = blockIdx.x;         // node
  int h = threadIdx.x;        // channel (CIN = 128 threads)
  float s = 0.0f; int c = 0;
  #pragma unroll
  for (int b = 0; b < BAG; ++b) {
    int ix = idx[i * BAG + b];
    if (ix != BF_DIM) { s += cb[(size_t)ix * CIN + h]; ++c; }
  }
  x[(size_t)i * CIN + h] = s / (float)(c > 0 ? c : 1);
}

// --------------------------------------------------------- graph preprocessing
__global__ void k_count_raw(const int* __restrict__ dst, int* __restrict__ cnt) {
  int e = blockIdx.x * blockDim.x + threadIdx.x;
  if (e < E_EDGES) atomicAdd(&cnt[dst[e]], 1);
}

__global__ void k_dedup(const int* __restrict__ src, const int* __restrict__ dst,
                        unsigned* __restrict__ bitmap, int* __restrict__ uqs,
                        int* __restrict__ uqd, int* __restrict__ nUnique,
                        int* __restrict__ cnt_ded, int* __restrict__ srccnt) {
  int e = blockIdx.x * blockDim.x + threadIdx.x;
  if (e >= E_EDGES) return;
  int s = src[e], d = dst[e];
  unsigned key = (unsigned)d * N_NODES + (unsigned)s;   // < 2^24
  unsigned bit = 1u << (key & 31u);
  unsigned old = atomicOr(&bitmap[key >> 5], bit);
  if (!(old & bit)) {
    int p = atomicAdd(nUnique, 1);
    uqs[p] = s; uqd[p] = d;
    atomicAdd(&cnt_ded[d], 1);
    atomicAdd(&srccnt[s], 1);
  }
}

// exclusive scan over exactly 4096 ints (1024 threads, 4 per thread)
__global__ void k_exscan4096(const int* __restrict__ cnt, int* __restrict__ rptr) {
  __shared__ int part[1024];
  int tid = threadIdx.x;
  int b = tid * 4;
  int a0 = cnt[b], a1 = cnt[b + 1], a2 = cnt[b + 2], a3 = cnt[b + 3];
  int s = a0 + a1 + a2 + a3;
  part[tid] = s;
  __syncthreads();
  for (int off = 1; off < 1024; off <<= 1) {
    int v = (tid >= off) ? part[tid - off] : 0;
    __syncthreads();
    part[tid] += v;
    __syncthreads();
  }
  int excl = part[tid] - s;
  rptr[b] = excl;
  rptr[b + 1] = excl + a0;
  rptr[b + 2] = excl + a0 + a1;
  rptr[b + 3] = excl + a0 + a1 + a2;
  if (tid == 1023) rptr[4096] = excl + s;
}

__global__ void k_fill_csr_raw(const int* __restrict__ src, const int* __restrict__ dst,
                               const int* __restrict__ rptr, int* __restrict__ fill,
                               int* __restrict__ col) {
  int e = blockIdx.x * blockDim.x + threadIdx.x;
  if (e >= E_EDGES) return;
  int d = dst[e];
  int p = atomicAdd(&fill[d], 1);
  col[rptr[d] + p] = src[e];
}

__global__ void k_fill_csr_ded(const int* __restrict__ uqs, const int* __restrict__ uqd,
                               const int* __restrict__ nU, const int* __restrict__ rptr,
                               int* __restrict__ fill, int* __restrict__ col) {
  int e = blockIdx.x * blockDim.x + threadIdx.x;
  if (e >= *nU) return;
  int d = uqd[e];
  int p = atomicAdd(&fill[d], 1);
  col[rptr[d] + p] = uqs[e];
}

__global__ void k_norms(const int* __restrict__ cnt_raw, const int* __restrict__ srccnt,
                        float* __restrict__ dis, float* __restrict__ di) {
  int i = blockIdx.x * blockDim.x + threadIdx.x;
  if (i >= N_NODES) return;
  dis[i] = rsqrtf((float)(cnt_raw[i] + 1));   // sparse-GCN deg (raw edges + self loop)
  di[i]  = rsqrtf((float)(srccnt[i] + 1));    // Ah row-sum (deduped out-degree + 1)
}

__global__ void k_dc(const float* __restrict__ w, float* __restrict__ dc) {
  int i = blockIdx.x * blockDim.x + threadIdx.x;
  if (i >= N_NODES) return;
  dc[i] = rsqrtf(fmaxf(w[i] + 1.0f, 1e-12f));
}

// ------------------------------------------------ Richardson step: V' = aU + b*AnT*V
__global__ void k_richardson(const float* __restrict__ Vin, const float* __restrict__ U,
                             float* __restrict__ Vout, const int* __restrict__ rptr,
                             const int* __restrict__ col, const float* __restrict__ di,
                             int H) {
  size_t t = (size_t)blockIdx.x * blockDim.x + threadIdx.x;
  if (t >= (size_t)N_NODES * H) return;
  int i = (int)(t / H);
  int h = (int)(t % H);
  float s = di[i] * Vin[t];                    // self-loop term of Ah^T
  int p1 = rptr[i + 1];
  for (int p = rptr[i]; p < p1; ++p) {
    int j = col[p];
    s += di[j] * Vin[(size_t)j * H + h];
  }
  Vout[t] = ALPHA_F * U[t] + BETA_F * di[i] * s;
}

// --------------------------------- sparse GCN gather: prelu(dis_i*(sum)+b, a)
__global__ void k_gcn_gather(const float* __restrict__ ZW, float* __restrict__ out,
                             const int* __restrict__ rptr, const int* __restrict__ col,
                             const float* __restrict__ dis, const float* __restrict__ bias,
                             const float* __restrict__ avec) {
  size_t t = (size_t)blockIdx.x * blockDim.x + threadIdx.x;
  if (t >= (size_t)N_NODES * HID) return;
  int i = (int)(t >> 9);
  int h = (int)(t & (HID - 1));
  float s = dis[i] * ZW[t];                    // self loop
  int p1 = rptr[i + 1];
  for (int p = rptr[i]; p < p1; ++p) {
    int j = col[p];
    s += dis[j] * ZW[(size_t)j * HID + h];
  }
  float v = dis[i] * s + bias[h];
  float a = avec[h];
  out[t] = (v >= 0.0f) ? v : a * v;
}

// ------------------------------------------------------------- elementwise ops
__global__ void k_rowscale(const float* __restrict__ Y, const float* __restrict__ dc,
                           float* __restrict__ U) {
  size_t t = (size_t)blockIdx.x * blockDim.x + threadIdx.x;
  if (t >= (size_t)N_NODES * HID) return;
  U[t] = dc[t >> 9] * Y[t];
}

__global__ void k_scale_alpha(const float* __restrict__ U, float* __restrict__ V) {
  size_t t = (size_t)blockIdx.x * blockDim.x + threadIdx.x;
  if (t >= (size_t)N_NODES * HID) return;
  V[t] = ALPHA_F * U[t];
}

__global__ void k_combine(const float* __restrict__ V, const float* __restrict__ U,
                          const float* __restrict__ dc, const float* __restrict__ bias,
                          const float* __restrict__ avec, float* __restrict__ out) {
  size_t t = (size_t)blockIdx.x * blockDim.x + threadIdx.x;
  if (t >= (size_t)N_NODES * HID) return;
  int i = (int)(t >> 9);
  int h = (int)(t & (HID - 1));
  float v = dc[i] * (V[t] + U[t]) + bias[h];
  float a = avec[h];
  out[t] = (v >= 0.0f) ? v : a * v;
}

__global__ void k_segsum(const float* __restrict__ z, const int* __restrict__ batch,
                         float* __restrict__ gv, int colOff) {
  size_t t = (size_t)blockIdx.x * blockDim.x + threadIdx.x;
  if (t >= (size_t)N_NODES * HID) return;
  int i = (int)(t >> 9);
  int h = (int)(t & (HID - 1));
  atomicAdd(&gv[(size_t)batch[i] * (2 * HID) + colOff + h], z[t]);
}

// ------------------------------------------------------------- WMMA fp32 GEMM
// C[M,N] = act(A[M,K] @ B[K,N] + bias) (+ addSrc). One wave per 16x16 tile.
// Lane layouts per CDNA5 ISA 7.12.2:
//   A 16x4 : lane&15 -> M, (lane>>4) selects K pair {0,1}/{2,3} across 2 VGPRs
//   B 4x16 : lane&15 -> N, (lane>>4) selects K pair across 2 VGPRs
//   C 16x16: vgpr r -> M = r + (lane>>4)*8, N = lane&15
__global__ void k_gemm_wmma(const float* __restrict__ A, const float* __restrict__ B,
                            float* __restrict__ C, int M, int N, int K,
                            const float* __restrict__ bias,
                            const float* __restrict__ alphaPtr, int alphaMode,
                            const float* __restrict__ addSrc) {
  const int lane = threadIdx.x & 31;
  const int wv   = threadIdx.x >> 5;
  const int tilesN = N >> 4;
  const int tile = blockIdx.x * 4 + wv;
  if (tile >= (M >> 4) * tilesN) return;
  const int tm = tile / tilesN, tn = tile % tilesN;
  const int half = lane >> 4;
  const int l    = lane & 15;
  const float* Arow = A + (size_t)(tm * 16 + l) * K;
  const float* Bcol = B + (size_t)(tn * 16 + l);
  v8f acc = {0.f, 0.f, 0.f, 0.f, 0.f, 0.f, 0.f, 0.f};
  for (int k0 = 0; k0 < K; k0 += 4) {
    v2f a, b;
    a[0] = Arow[k0 + half * 2 + 0];
    a[1] = Arow[k0 + half * 2 + 1];
    b[0] = Bcol[(size_t)(k0 + half * 2 + 0) * N];
    b[1] = Bcol[(size_t)(k0 + half * 2 + 1) * N];
    acc = __builtin_amdgcn_wmma_f32_16x16x4_f32(false, a, false, b, (short)0, acc,
                                                false, false);
  }
  const int colg = tn * 16 + l;
  const float bval = bias ? bias[colg] : 0.0f;
  const float aval = (alphaMode == 1) ? alphaPtr[0] : 0.0f;
  #pragma unroll
  for (int r = 0; r < 8; ++r) {
    int rowg = tm * 16 + half * 8 + r;
    float v = acc[r] + bval;
    if (alphaMode == 1) v = (v >= 0.0f) ? v : aval * v;
    if (addSrc) v += addSrc[(size_t)rowg * N + colg];
    C[(size_t)rowg * N + colg] = v;
  }
}

// ---------------------------------------------------------------------- launch
extern "C" void kernel_launch(void* const* d_in, const int* in_sizes, int n_in,
                              void* d_out, int out_size, void* d_ws, size_t ws_size,
                              hipStream_t stream) {
  const float* codebook = (const float*)d_in[0];
  const float* g1_W0 = (const float*)d_in[1];  const float* g1_b0 = (const float*)d_in[2];
  const float* g1_W1 = (const float*)d_in[3];  const float* g1_b1 = (const float*)d_in[4];
  const float* g1_a  = (const float*)d_in[5];
  const float* g2_W0 = (const float*)d_in[6];  const float* g2_b0 = (const float*)d_in[7];
  const float* g2_W1 = (const float*)d_in[8];  const float* g2_b1 = (const float*)d_in[9];
  const float* g2_a  = (const float*)d_in[10];
  const float* m1_W1 = (const float*)d_in[11]; const float* m1_b1 = (const float*)d_in[12];
  const float* m1_W2 = (const float*)d_in[13]; const float* m1_b2 = (const float*)d_in[14];
  const float* m1_W3 = (const float*)d_in[15]; const float* m1_b3 = (const float*)d_in[16];
  const float* m1_a1 = (const float*)d_in[17]; const float* m1_a2 = (const float*)d_in[18];
  const float* m1_a3 = (const float*)d_in[19];
  const float* m1_Ws = (const float*)d_in[20]; const float* m1_bs = (const float*)d_in[21];
  const float* m2_W1 = (const float*)d_in[22]; const float* m2_b1 = (const float*)d_in[23];
  const float* m2_W2 = (const float*)d_in[24]; const float* m2_b2 = (const float*)d_in[25];
  const float* m2_W3 = (const float*)d_in[26]; const float* m2_b3 = (const float*)d_in[27];
  const float* m2_a1 = (const float*)d_in[28]; const float* m2_a2 = (const float*)d_in[29];
  const float* m2_a3 = (const float*)d_in[30];
  const float* m2_Ws = (const float*)d_in[31]; const float* m2_bs = (const float*)d_in[32];
  const int* x_idx = (const int*)d_in[33];
  const int* edge  = (const int*)d_in[34];
  const int* batch = (const int*)d_in[35];
  const int* e_src = edge;
  const int* e_dst = edge + E_EDGES;

  // --- workspace carve (256B aligned) ---
  char* w = (char*)d_ws;
  size_t off = 0;
  auto carveF = [&](size_t n) -> float* { float* p = (float*)(w + off); off += ((n * 4 + 255) / 256) * 256; return p; };
  auto carveI = [&](size_t n) -> int*   { int*   p = (int*)(w + off);   off += ((n * 4 + 255) / 256) * 256; return p; };

  const size_t NH = (size_t)N_NODES * HID;
  float* x  = carveF((size_t)N_NODES * CIN);
  float* B0 = carveF(NH); float* B1 = carveF(NH); float* B2 = carveF(NH);
  float* B3 = carveF(NH); float* B4 = carveF(NH); float* B5 = carveF(NH);
  float* Gc1 = carveF((size_t)G_GRAPHS * 2 * HID);
  float* Gc2 = carveF((size_t)G_GRAPHS * 2 * HID);
  float* Gt1 = carveF((size_t)G_GRAPHS * HID);
  float* Gt2 = carveF((size_t)G_GRAPHS * HID);
  float* dis = carveF(N_NODES); float* di = carveF(N_NODES); float* dc = carveF(N_NODES);
  float* wU  = carveF(N_NODES); float* wA = carveF(N_NODES); float* wB = carveF(N_NODES);
  int* cnt_raw = carveI(N_NODES); int* cnt_ded = carveI(N_NODES); int* srccnt = carveI(N_NODES);
  int* fill1 = carveI(N_NODES);   int* fill2 = carveI(N_NODES);
  int* rptr_raw = carveI(N_NODES + 1); int* rptr_ded = carveI(N_NODES + 1);
  int* col_raw = carveI(E_EDGES); int* uqs = carveI(E_EDGES);
  int* uqd = carveI(E_EDGES);     int* col_ded = carveI(E_EDGES);
  int* nUnique = carveI(1);
  unsigned* bitmap = (unsigned*)carveI((size_t)N_NODES * N_NODES / 32);

  float* lv1  = (float*)d_out;
  float* gv1o = lv1 + NH;
  float* lv2  = gv1o + (size_t)G_GRAPHS * HID;
  float* gv2o = lv2 + NH;

  auto nb = [](size_t n, int b) { return (int)((n + b - 1) / b); };
  const int B = 256;
  const int gNH = nb(NH, B);
  const int tilesNH = (N_NODES / 16) * (HID / 16);     // 8192 tiles
  const int tilesG  = (G_GRAPHS / 16) * (HID / 16);    // 128 tiles

  // --- init / zero (must happen every call: graph is replayed) ---
  k_fill_i<<<nb((size_t)N_NODES * N_NODES / 32, B), B, 0, stream>>>((int*)bitmap, (long)N_NODES * N_NODES / 32, 0);
  k_fill_i<<<nb(N_NODES, B), B, 0, stream>>>(cnt_raw, N_NODES, 0);
  k_fill_i<<<nb(N_NODES, B), B, 0, stream>>>(cnt_ded, N_NODES, 0);
  k_fill_i<<<nb(N_NODES, B), B, 0, stream>>>(srccnt, N_NODES, 0);
  k_fill_i<<<nb(N_NODES, B), B, 0, stream>>>(fill1, N_NODES, 0);
  k_fill_i<<<nb(N_NODES, B), B, 0, stream>>>(fill2, N_NODES, 0);
  k_fill_i<<<1, 32, 0, stream>>>(nUnique, 1, 0);
  k_fill_f<<<nb((size_t)G_GRAPHS * 2 * HID, B), B, 0, stream>>>(Gc1, (long)G_GRAPHS * 2 * HID, 0.0f);
  k_fill_f<<<nb((size_t)G_GRAPHS * 2 * HID, B), B, 0, stream>>>(Gc2, (long)G_GRAPHS * 2 * HID, 0.0f);
  k_fill_f<<<nb(N_NODES, B), B, 0, stream>>>(wU, N_NODES, 1.0f);
  k_fill_f<<<nb(N_NODES, B), B, 0, stream>>>(wA, N_NODES, ALPHA_F);

  // --- features + graph structure ---
  k_embed<<<N_NODES, CIN, 0, stream>>>(codebook, x_idx, x);
  k_count_raw<<<nb(E_EDGES, B), B, 0, stream>>>(e_dst, cnt_raw);
  k_dedup<<<nb(E_EDGES, B), B, 0, stream>>>(e_src, e_dst, bitmap, uqs, uqd, nUnique, cnt_ded, srccnt);
  k_exscan4096<<<1, 1024, 0, stream>>>(cnt_raw, rptr_raw);
  k_exscan4096<<<1, 1024, 0, stream>>>(cnt_ded, rptr_ded);
  k_fill_csr_raw<<<nb(E_EDGES, B), B, 0, stream>>>(e_src, e_dst, rptr_raw, fill1, col_raw);
  k_fill_csr_ded<<<nb(E_EDGES, B), B, 0, stream>>>(uqs, uqd, nUnique, rptr_ded, fill2, col_ded);
  k_norms<<<nb(N_NODES, B), B, 0, stream>>>(cnt_raw, srccnt, dis, di);

  // --- column-sum solve: w = S^T 1 via Richardson on M^T (64 iters, rate 0.8) ---
  {
    float* va = wA; float* vb = wB;
    for (int it = 0; it < N_ITERS; ++it) {
      k_richardson<<<nb(N_NODES, B), B, 0, stream>>>(va, wU, vb, rptr_ded, col_ded, di, 1);
      float* t = va; va = vb; vb = t;
    }
    k_dc<<<nb(N_NODES, B), B, 0, stream>>>(va, dc);   // va == wA after even #swaps
  }

  // ================= Tower 1 (sparse GCN, raw edges incl. duplicates) =========
  k_gemm_wmma<<<nb(tilesNH, 4), 128, 0, stream>>>(x, g1_W0, B0, N_NODES, HID, CIN, nullptr, nullptr, 0, nullptr);
  k_gcn_gather<<<gNH, B, 0, stream>>>(B0, B1, rptr_raw, col_raw, dis, g1_b0, g1_a);  // z1
  k_gemm_wmma<<<nb(tilesNH, 4), 128, 0, stream>>>(B1, g1_W1, B0, N_NODES, HID, HID, nullptr, nullptr, 0, nullptr);
  k_gcn_gather<<<gNH, B, 0, stream>>>(B0, B2, rptr_raw, col_raw, dis, g1_b1, g1_a);  // z2
  k_segsum<<<gNH, B, 0, stream>>>(B1, batch, Gc1, 0);
  k_segsum<<<gNH, B, 0, stream>>>(B2, batch, Gc1, HID);

  // MLP1(z2) -> lv1   (z2 in B2)
  k_gemm_wmma<<<nb(tilesNH, 4), 128, 0, stream>>>(B2, m1_W1, B0, N_NODES, HID, HID, m1_b1, m1_a1, 1, nullptr);
  k_gemm_wmma<<<nb(tilesNH, 4), 128, 0, stream>>>(B0, m1_W2, B1, N_NODES, HID, HID, m1_b2, m1_a2, 1, nullptr);
  k_gemm_wmma<<<nb(tilesNH, 4), 128, 0, stream>>>(B1, m1_W3, B3, N_NODES, HID, HID, m1_b3, m1_a3, 1, nullptr);
  k_gemm_wmma<<<nb(tilesNH, 4), 128, 0, stream>>>(B2, m1_Ws, lv1, N_NODES, HID, HID, m1_bs, nullptr, 0, B3);

  // ================= Tower 2 (PPR diffusion via sparse transpose solves) ======
  // layer 1
  k_gemm_wmma<<<nb(tilesNH, 4), 128, 0, stream>>>(x, g2_W0, B0, N_NODES, HID, CIN, nullptr, nullptr, 0, nullptr);
  k_rowscale<<<gNH, B, 0, stream>>>(B0, dc, B3);          // U = dc .* (x@W0)
  k_scale_alpha<<<gNH, B, 0, stream>>>(B3, B4);           // V0 = alpha*U
  {
    float* va = B4; float* vb = B5;
    for (int it = 0; it < N_ITERS; ++it) {
      k_richardson<<<gNH, B, 0, stream>>>(va, B3, vb, rptr_ded, col_ded, di, HID);
      float* t = va; va = vb; vb = t;
    }
  }
  k_combine<<<gNH, B, 0, stream>>>(B4, B3, dc, g2_b0, g2_a, B1);   // z1d
  k_segsum<<<gNH, B, 0, stream>>>(B1, batch, Gc2, 0);
  // layer 2
  k_gemm_wmma<<<nb(tilesNH, 4), 128, 0, stream>>>(B1, g2_W1, B0, N_NODES, HID, HID, nullptr, nullptr, 0, nullptr);
  k_rowscale<<<gNH, B, 0, stream>>>(B0, dc, B3);
  k_scale_alpha<<<gNH, B, 0, stream>>>(B3, B4);
  {
    float* va = B4; float* vb = B5;
    for (int it = 0; it < N_ITERS; ++it) {
      k_richardson<<<gNH, B, 0, stream>>>(va, B3, vb, rptr_ded, col_ded, di, HID);
      float* t = va; va = vb; vb = t;
    }
  }
  k_combine<<<gNH, B, 0, stream>>>(B4, B3, dc, g2_b1, g2_a, B2);   // z2d
  k_segsum<<<gNH, B, 0, stream>>>(B2, batch, Gc2, HID);

  // MLP1(z2d) -> lv2   (z2d in B2)
  k_gemm_wmma<<<nb(tilesNH, 4), 128, 0, stream>>>(B2, m1_W1, B0, N_NODES, HID, HID, m1_b1, m1_a1, 1, nullptr);
  k_gemm_wmma<<<nb(tilesNH, 4), 128, 0, stream>>>(B0, m1_W2, B1, N_NODES, HID, HID, m1_b2, m1_a2, 1, nullptr);
  k_gemm_wmma<<<nb(tilesNH, 4), 128, 0, stream>>>(B1, m1_W3, B3, N_NODES, HID, HID, m1_b3, m1_a3, 1, nullptr);
  k_gemm_wmma<<<nb(tilesNH, 4), 128, 0, stream>>>(B2, m1_Ws, lv2, N_NODES, HID, HID, m1_bs, nullptr, 0, B3);

  // ================= MLP2 on graph readouts [64, 1024] -> [64, 512] ===========
  k_gemm_wmma<<<nb(tilesG, 4), 128, 0, stream>>>(Gc1, m2_W1, Gt1, G_GRAPHS, HID, 2 * HID, m2_b1, m2_a1, 1, nullptr);
  k_gemm_wmma<<<nb(tilesG, 4), 128, 0, stream>>>(Gt1, m2_W2, Gt2, G_GRAPHS, HID, HID, m2_b2, m2_a2, 1, nullptr);
  k_gemm_wmma<<<nb(tilesG, 4), 128, 0, stream>>>(Gt2, m2_W3, Gt1, G_GRAPHS, HID, HID, m2_b3, m2_a3, 1, nullptr);
  k_gemm_wmma<<<nb(tilesG, 4), 128, 0, stream>>>(Gc1, m2_Ws, gv1o, G_GRAPHS, HID, 2 * HID, m2_bs, nullptr, 0, Gt1);

  k_gemm_wmma<<<nb(tilesG, 4), 128, 0, stream>>>(Gc2, m2_W1, Gt1, G_GRAPHS, HID, 2 * HID, m2_b1, m2_a1, 1, nullptr);
  k_gemm_wmma<<<nb(tilesG, 4), 128, 0, stream>>>(Gt1, m2_W2, Gt2, G_GRAPHS, HID, HID, m2_b2, m2_a2, 1, nullptr);
  k_gemm_wmma<<<nb(tilesG, 4), 128, 0, stream>>>(Gt2, m2_W3, Gt1, G_GRAPHS, HID, HID, m2_b3, m2_a3, 1, nullptr);
  k_gemm_wmma<<<nb(tilesG, 4), 128, 0, stream>>>(Gc2, m2_Ws, gv2o, G_GRAPHS, HID, 2 * HID, m2_bs, nullptr, 0, Gt1);
}